// StableSystem_48515950576192
// MI455X (gfx1250) — compile-verified
//
#include <hip/hip_runtime.h>
#include <hip/hip_bf16.h>

// Problem constants (from reference): NX=128, NU=64, BATCH=2048, T=1024, H=0.01
#define NX   128
#define NU   64
#define HSTEP 0.01f

#define XSTR 132            // x LDS row stride (floats), 132 % 64 == 4 -> conflict-free A frags
#define XSLOT (16 * XSTR)   // one x tile slot
#define USTR 68             // u LDS row stride (floats), 68 % 64 == 4
#define USLOT (16 * USTR)   // one u tile slot (16 batch rows)

#define MT_PACKED_DWORDS (8 * 16 * 32 * 4)   // 8 ntiles * 16 K-pairs * 32 lanes * 4 dwords = 16384
#define BT_PACKED_DWORDS (8 * 8  * 32 * 4)   // 8 ntiles *  8 K-pairs * 32 lanes * 4 dwords =  8192
#define PACKED_TOTAL     (MT_PACKED_DWORDS + BT_PACKED_DWORDS)  // 24576 dwords = 96KB

typedef __attribute__((ext_vector_type(2))) float v2f;
typedef __attribute__((ext_vector_type(4))) float v4f;
typedef __attribute__((ext_vector_type(8))) float v8f;

__device__ __forceinline__ v8f wmma4(v2f a, v2f b, v8f c) {
    // D = A(16x4 f32) * B(4x16 f32) + C(16x16 f32)
    return __builtin_amdgcn_wmma_f32_16x16x4_f32(
        /*neg_a=*/false, a, /*neg_b=*/false, b,
        /*c_mod=*/(short)0, c, /*reuse_a=*/false, /*reuse_b=*/false);
}

// ---------------------------------------------------------------------------
// Prep: build MT = I + h*A^T and BTh = h*B^T directly in WMMA B-fragment
// packed layout: for K-pair p (K in [8p, 8p+8)), lane = l + 16*hi (l = N&15),
// q in 0..3 holds K = 8p + 4*(q>>1) + 2*hi + (q&1). One v4f per (ntile,p,lane).
// ---------------------------------------------------------------------------
__global__ void prep_kernel(const float* __restrict__ A,
                            const float* __restrict__ B,
                            float* __restrict__ ws) {
    int i = blockIdx.x * blockDim.x + threadIdx.x;
    if (i < MT_PACKED_DWORDS) {
        int K = i >> 7, N = i & 127;                               // MT [128 x 128]
        float v = (K == N ? 1.0f : 0.0f) + HSTEP * A[N * NX + K];  // I + h*A^T
        int nt = N >> 4, l = N & 15;
        int p = K >> 3, r3 = K & 7;
        int hi = (r3 >> 1) & 1;
        int q = (r3 >> 2) * 2 + (r3 & 1);
        int lane = l + 16 * hi;
        ws[((nt * 16 + p) * 32 + lane) * 4 + q] = v;
    } else if (i < PACKED_TOTAL) {
        int j = i - MT_PACKED_DWORDS;
        int K = j >> 7, N = j & 127;                               // BTh [64 x 128]
        float v = HSTEP * B[N * NU + K];                           // h*B^T
        int nt = N >> 4, l = N & 15;
        int p = K >> 3, r3 = K & 7;
        int hi = (r3 >> 1) & 1;
        int q = (r3 >> 2) * 2 + (r3 & 1);
        int lane = l + 16 * hi;
        ws[MT_PACKED_DWORDS + ((nt * 8 + p) * 32 + lane) * 4 + q] = v;
    }
}

// ---------------------------------------------------------------------------
// Rollout: one block (8 waves, 256 threads) per 16-row batch tile.
// Each wave owns ONE 16-column output tile (ntile = wave id) and keeps ALL
// of its MT/BTh WMMA B-fragments resident in VGPRs for the whole rollout
// (96 VGPRs, loaded once). Per step: 48 conflict-free ds_load_b64 A-frags
// (paired into ds_load_2addr_b64 by the compiler), 48 v_wmma_f32_16x16x4_f32
// in TWO independent accumulation chains, 8 LDS + 8 global stores.
// x tile is double-buffered in LDS -> single barrier per step.
// ---------------------------------------------------------------------------
__global__ __launch_bounds__(256, 1)
void rollout_kernel(const float* __restrict__ xi,
                    const float* __restrict__ u,
                    const float* __restrict__ packed,
                    float* __restrict__ out,
                    int T) {
    extern __shared__ float sm[];
    float* xb = sm;                      // [2 * XSLOT] double-buffered x tile
    float* ub = sm + 2 * XSLOT;          // [2 * USLOT] double-buffered u tile

    const int tid  = threadIdx.x;
    const int wv   = tid >> 5;           // wave id == output ntile
    const int lane = tid & 31;
    const int l    = lane & 15;
    const int hi   = lane >> 4;
    const int brow = blockIdx.x * 16;    // batch tile base row

    // Load this wave's loop-invariant B-fragments into registers (held all T steps).
    v4f bMT[16];
    v4f bBT[8];
#pragma unroll
    for (int p = 0; p < 16; ++p)
        bMT[p] = *(const v4f*)(packed + ((wv * 16 + p) * 32 + lane) * 4);
#pragma unroll
    for (int p = 0; p < 8; ++p)
        bBT[p] = *(const v4f*)(packed + MT_PACKED_DWORDS + ((wv * 8 + p) * 32 + lane) * 4);

    // x0 -> LDS slot 0 (row-major, stride XSTR) and output t-index 0.
    for (int i = tid; i < 16 * 32; i += 256) {
        int row = i >> 5, c4 = i & 31;
        float4 v = *(const float4*)(xi + (size_t)(brow + row) * NX + c4 * 4);
        *(float4*)(xb + row * XSTR + c4 * 4) = v;
        *(float4*)(out + ((size_t)(brow + row) * T) * NX + c4 * 4) = v;
    }

    // u tile: each thread owns exactly one float4 (16 rows x 16 float4s).
    const int urow = tid >> 4;
    const int uc4  = tid & 15;
    float4 ur = *(const float4*)(u + ((size_t)(brow + urow) * T + 0) * NU + uc4 * 4);
    *(float4*)(ub + 0 + urow * USTR + uc4 * 4) = ur;

    __syncthreads();

    for (int t = 0; t < T; ++t) {
        const int cur  = (t & 1);
        const float* xr = xb + cur * XSLOT;            // read slot
        float*       xw = xb + (1 - cur) * XSLOT;      // write slot
        const float* uR = ub + cur * USLOT;

        if (t + 1 < T)   // prefetch next u behind this step's WMMAs
            ur = *(const float4*)(u + ((size_t)(brow + urow) * T + (t + 1)) * NU + uc4 * 4);

        // Two independent accumulation chains to keep the matrix pipe full.
        v8f cA = {};
        v8f cB = {};

        // x_t @ MT : 16 K-pairs (K = 128), B operands from registers.
#pragma unroll
        for (int p = 0; p < 16; ++p) {
            v2f a0 = *(const v2f*)(xr + l * XSTR + p * 8 + 2 * hi);
            v2f a1 = *(const v2f*)(xr + l * XSTR + p * 8 + 4 + 2 * hi);
            cA = wmma4(a0, __builtin_shufflevector(bMT[p], bMT[p], 0, 1), cA);
            cB = wmma4(a1, __builtin_shufflevector(bMT[p], bMT[p], 2, 3), cB);
        }
        // u_t @ BTh : 8 K-pairs (K = 64).
#pragma unroll
        for (int p = 0; p < 8; ++p) {
            v2f a0 = *(const v2f*)(uR + l * USTR + p * 8 + 2 * hi);
            v2f a1 = *(const v2f*)(uR + l * USTR + p * 8 + 4 + 2 * hi);
            cA = wmma4(a0, __builtin_shufflevector(bBT[p], bBT[p], 0, 1), cA);
            cB = wmma4(a1, __builtin_shufflevector(bBT[p], bBT[p], 2, 3), cB);
        }

        v8f c = cA + cB;

        // Write x_{t+1} column block into the OTHER x slot (no WAR with readers).
#pragma unroll
        for (int r = 0; r < 8; ++r)
            xw[(r + 8 * hi) * XSTR + wv * 16 + l] = c[r];

        // Stage prefetched u(t+1) into the other u slot.
        if (t + 1 < T)
            *(float4*)(ub + (1 - cur) * USLOT + urow * USTR + uc4 * 4) = ur;

        // Output: x_{t+1} at time index t for t>=1 (x1 dropped by reference).
        if (t >= 1) {
#pragma unroll
            for (int r = 0; r < 8; ++r) {
                int M = r + 8 * hi;
                out[((size_t)(brow + M) * T + t) * NX + wv * 16 + l] = c[r];
            }
        }

        __syncthreads();   // xb[next] / ub[next] fully written before next step reads
    }
}

extern "C" void kernel_launch(void* const* d_in, const int* in_sizes, int n_in,
                              void* d_out, int out_size, void* d_ws, size_t ws_size,
                              hipStream_t stream) {
    const float* xi = (const float*)d_in[0];   // [BATCH, 1, NX]
    const float* u  = (const float*)d_in[1];   // [BATCH, T, NU]
    const float* A  = (const float*)d_in[2];   // [NX, NX]
    const float* B  = (const float*)d_in[3];   // [NX, NU]

    const int batch = in_sizes[0] / NX;
    const int T     = in_sizes[1] / (batch * NU);

    float* ws = (float*)d_ws;  // needs PACKED_TOTAL*4 = 96KB

    prep_kernel<<<(PACKED_TOTAL + 255) / 256, 256, 0, stream>>>(A, B, ws);

    const size_t smem = (size_t)(2 * XSLOT + 2 * USLOT) * 4;   // 25,600 B
    rollout_kernel<<<dim3(batch / 16), dim3(256), smem, stream>>>(
        xi, u, ws, (float*)d_out, T);
}